// ConvBlock_32375463478029
// MI455X (gfx1250) — compile-verified
//
#include <hip/hip_runtime.h>
#include <math.h>

#define EPS_MSG 1e-7f
#define EPS_BN  1e-5f

typedef __attribute__((ext_vector_type(2))) float v2f;
typedef __attribute__((ext_vector_type(8))) float v8f;

// ---------------------------------------------------------------- zero
__global__ void zero_f4(float4* __restrict__ p, int n4) {
    int i = blockIdx.x * blockDim.x + threadIdx.x;
    if (i < n4) p[i] = make_float4(0.f, 0.f, 0.f, 0.f);
}

// ---------------------------------------------------------------- edge pass 1: segment max via uint atomicMax (msg > 0)
__global__ void edge_max_kernel(const float* __restrict__ x, const int* __restrict__ ei,
                                const float* __restrict__ ea, unsigned* __restrict__ mx,
                                int E, int total) {
    int i = blockIdx.x * blockDim.x + threadIdx.x;
    if (i >= total) return;
    int e  = i >> 4;
    int c4 = (i & 15) << 2;
    int src = ei[e];
    int dst = ei[E + e];
    float4 xv = *reinterpret_cast<const float4*>(x  + (size_t)src * 64 + c4);
    float4 av = *reinterpret_cast<const float4*>(ea + (size_t)e   * 64 + c4);
    unsigned* m = mx + (size_t)dst * 64 + c4;
    float m0 = fmaxf(xv.x + av.x, 0.f) + EPS_MSG;
    float m1 = fmaxf(xv.y + av.y, 0.f) + EPS_MSG;
    float m2 = fmaxf(xv.z + av.z, 0.f) + EPS_MSG;
    float m3 = fmaxf(xv.w + av.w, 0.f) + EPS_MSG;
    atomicMax(m + 0, __float_as_uint(m0));
    atomicMax(m + 1, __float_as_uint(m1));
    atomicMax(m + 2, __float_as_uint(m2));
    atomicMax(m + 3, __float_as_uint(m3));
}

// ---------------------------------------------------------------- edge pass 2: accumulate den = sum(exp), num = sum(msg*exp)
__global__ void edge_sum_kernel(const float* __restrict__ x, const int* __restrict__ ei,
                                const float* __restrict__ ea, const float* __restrict__ mx,
                                float* __restrict__ den, float* __restrict__ num,
                                int E, int total) {
    int i = blockIdx.x * blockDim.x + threadIdx.x;
    if (i >= total) return;
    int e  = i >> 4;
    int c4 = (i & 15) << 2;
    int src = ei[e];
    int dst = ei[E + e];
    float4 xv = *reinterpret_cast<const float4*>(x  + (size_t)src * 64 + c4);
    float4 av = *reinterpret_cast<const float4*>(ea + (size_t)e   * 64 + c4);
    float4 mv = *reinterpret_cast<const float4*>(mx + (size_t)dst * 64 + c4);
    float m0 = fmaxf(xv.x + av.x, 0.f) + EPS_MSG;
    float m1 = fmaxf(xv.y + av.y, 0.f) + EPS_MSG;
    float m2 = fmaxf(xv.z + av.z, 0.f) + EPS_MSG;
    float m3 = fmaxf(xv.w + av.w, 0.f) + EPS_MSG;
    float e0 = __expf(m0 - mv.x);
    float e1 = __expf(m1 - mv.y);
    float e2 = __expf(m2 - mv.z);
    float e3 = __expf(m3 - mv.w);
    float* dp = den + (size_t)dst * 64 + c4;
    float* np = num + (size_t)dst * 64 + c4;
    atomicAdd(dp + 0, e0); atomicAdd(np + 0, m0 * e0);
    atomicAdd(dp + 1, e1); atomicAdd(np + 1, m1 * e1);
    atomicAdd(dp + 2, e2); atomicAdd(np + 2, m2 * e2);
    atomicAdd(dp + 3, e3); atomicAdd(np + 3, m3 * e3);
}

// ---------------------------------------------------------------- h0 = num/(den+1e-16) + x
__global__ void combine_kernel(const float4* __restrict__ num, const float4* __restrict__ den,
                               const float4* __restrict__ x, float4* __restrict__ h0, int n4) {
    int i = blockIdx.x * blockDim.x + threadIdx.x;
    if (i >= n4) return;
    float4 nv = num[i], dv = den[i], xv = x[i];
    float4 r;
    r.x = fmaf(nv.x, __builtin_amdgcn_rcpf(dv.x + 1e-16f), xv.x);
    r.y = fmaf(nv.y, __builtin_amdgcn_rcpf(dv.y + 1e-16f), xv.y);
    r.z = fmaf(nv.z, __builtin_amdgcn_rcpf(dv.z + 1e-16f), xv.z);
    r.w = fmaf(nv.w, __builtin_amdgcn_rcpf(dv.w + 1e-16f), xv.w);
    h0[i] = r;
}

// ---------------------------------------------------------------- WMMA f32 GEMM, one wave = one 16x16 tile of D = act(A)*B (+bias)
// MODE 0: A plain; MODE 1: relu(A*scale+shift); MODE 2: silu(A*scale+shift)
// K and NCOLS are compile-time so all global addresses are base + immediate offset.
template <int K, int NCOLS, int MODE, bool HAS_BIAS>
__global__ void __launch_bounds__(32)
gemm16_wmma(const float* __restrict__ A, const float* __restrict__ B, float* __restrict__ D,
            const float* __restrict__ bias, const float* __restrict__ scale,
            const float* __restrict__ shift) {
    int n0   = blockIdx.x << 4;
    int m0   = blockIdx.y << 4;
    int lane = threadIdx.x;
    int half = lane >> 4;       // 0: K pair {0,1}; 1: K pair {2,3}
    int l16  = lane & 15;
    int koff = half << 1;       // 0 or 2
    const float* Arow  = A + (size_t)(m0 + l16) * K + koff;   // + kb gives immediate offsets
    const float* Bcol  = B + (size_t)koff * NCOLS + n0 + l16; // + kb*NCOLS gives imm offsets
    const float* scp   = (MODE != 0) ? (scale + koff) : nullptr;
    const float* shp   = (MODE != 0) ? (shift + koff) : nullptr;
    v8f acc = {0.f, 0.f, 0.f, 0.f, 0.f, 0.f, 0.f, 0.f};
#pragma unroll
    for (int kb = 0; kb < K; kb += 4) {
        v2f av = *reinterpret_cast<const v2f*>(Arow + kb);    // one b64 load
        if (MODE == 1) {
            v2f sc = *reinterpret_cast<const v2f*>(scp + kb);
            v2f sh = *reinterpret_cast<const v2f*>(shp + kb);
            av.x = fmaxf(fmaf(av.x, sc.x, sh.x), 0.f);
            av.y = fmaxf(fmaf(av.y, sc.y, sh.y), 0.f);
        } else if (MODE == 2) {
            v2f sc = *reinterpret_cast<const v2f*>(scp + kb);
            v2f sh = *reinterpret_cast<const v2f*>(shp + kb);
            float u0 = fmaf(av.x, sc.x, sh.x);
            float u1 = fmaf(av.y, sc.y, sh.y);
            av.x = u0 * __builtin_amdgcn_rcpf(1.f + __expf(-u0));
            av.y = u1 * __builtin_amdgcn_rcpf(1.f + __expf(-u1));
        }
        v2f bv = {Bcol[(size_t)kb * NCOLS],
                  Bcol[(size_t)(kb + 1) * NCOLS]};
        // D(16x16,f32) += A(16x4,f32) x B(4x16,f32)
        acc = __builtin_amdgcn_wmma_f32_16x16x4_f32(false, av, false, bv,
                                                    (short)0, acc, false, false);
    }
    float bc = HAS_BIAS ? bias[n0 + l16] : 0.f;
    float* Dp = D + ((size_t)m0 + (half << 3)) * NCOLS + n0 + l16;
#pragma unroll
    for (int v = 0; v < 8; ++v) {
        Dp[(size_t)v * NCOLS] = acc[v] + bc;   // row = v + 8*half (imm offsets)
    }
}

// ---------------------------------------------------------------- per-column sum / sumsq (for training-mode BN)
__global__ void stats_kernel(const float* __restrict__ src, int C, int Nrows,
                             float* __restrict__ sums) {
    int t   = blockIdx.x * blockDim.x + threadIdx.x;
    int col = t % C;
    int r0  = t / C;
    int rstride = (gridDim.x * blockDim.x) / C;
    float s = 0.f, s2 = 0.f;
    for (int r = r0; r < Nrows; r += rstride) {
        float v = src[(size_t)r * C + col];
        s += v; s2 += v * v;
    }
    atomicAdd(&sums[col], s);
    atomicAdd(&sums[C + col], s2);
}

// ---------------------------------------------------------------- BN scale/shift: scale=g*rsqrt(var+eps), shift=be-mean*scale
__global__ void finalize_kernel(const float* __restrict__ sums, const float* __restrict__ g,
                                const float* __restrict__ be, float* __restrict__ scale,
                                float* __restrict__ shift, int C, float invN) {
    int c = threadIdx.x;
    if (c >= C) return;
    float mean = sums[c] * invN;
    float var  = sums[C + c] * invN - mean * mean;
    float sc   = g[c] * rsqrtf(var + EPS_BN);
    scale[c] = sc;
    shift[c] = be[c] - mean * sc;
}

// ---------------------------------------------------------------- out = silu(h*scale+shift)
__global__ void bn_silu_out(const float* __restrict__ h, const float* __restrict__ scale,
                            const float* __restrict__ shift, float* __restrict__ out, int total) {
    int i = blockIdx.x * blockDim.x + threadIdx.x;
    if (i >= total) return;
    int c = i & 63;
    float u = fmaf(h[i], scale[c], shift[c]);
    out[i] = u * __builtin_amdgcn_rcpf(1.f + __expf(-u));
}

// ================================================================ launch
extern "C" void kernel_launch(void* const* d_in, const int* in_sizes, int n_in,
                              void* d_out, int out_size, void* d_ws, size_t ws_size,
                              hipStream_t stream) {
    const float* x      = (const float*)d_in[0];
    const int*   ei     = (const int*)  d_in[1];
    const float* ea     = (const float*)d_in[2];
    /* d_in[3] = pos, unused by reference */
    const float* W1     = (const float*)d_in[4];
    const float* b1     = (const float*)d_in[5];
    const float* g_mlp  = (const float*)d_in[6];
    const float* be_mlp = (const float*)d_in[7];
    const float* W2     = (const float*)d_in[8];
    const float* b2     = (const float*)d_in[9];
    const float* g1     = (const float*)d_in[10];
    const float* be1    = (const float*)d_in[11];
    const float* Wl     = (const float*)d_in[12];
    const float* g2     = (const float*)d_in[13];
    const float* be2    = (const float*)d_in[14];

    const int N = in_sizes[0] / 64;        // 100000 (divisible by 16)
    const int E = in_sizes[1] / 2;         // 1600000
    float* out = (float*)d_out;

    float*  ws   = (float*)d_ws;
    size_t  NC   = (size_t)N * 64;
    float*  mxb  = ws;                      // [N,64] msg max  -> reused as h0
    float*  denb = ws + NC;                 // [N,64] sum(exp) -> reused as h2
    float*  numb = ws + 2 * NC;             // [N,64] sum(m*e) -> reused as h3
    float*  t1   = ws + 3 * NC;             // [N,128]
    float*  st   = t1 + (size_t)N * 128;    // 1024 floats of stats/scales
    // stats layout (floats): [0]sum1/sumsq1(256) [256]sum2/sumsq2(128) [384]sum3/sumsq3(128)
    //                        [512]scale1(128) [640]shift1(128) [768]scaleA [832]shiftA [896]scaleB [960]shiftB

    const int TB = 256;
    // zero mx/den/num + stats
    {
        int n4 = (int)(3 * NC / 4);
        zero_f4<<<(n4 + TB - 1) / TB, TB, 0, stream>>>((float4*)mxb, n4);
        zero_f4<<<1, TB, 0, stream>>>((float4*)st, 256);
    }
    // edge phase
    {
        int total = E * 16;
        int nb = (total + TB - 1) / TB;
        edge_max_kernel<<<nb, TB, 0, stream>>>(x, ei, ea, (unsigned*)mxb, E, total);
        edge_sum_kernel<<<nb, TB, 0, stream>>>(x, ei, ea, mxb, denb, numb, E, total);
    }
    // h0 = num/(den+1e-16) + x  (overwrites mx buffer)
    {
        int n4 = (int)(NC / 4);
        combine_kernel<<<(n4 + TB - 1) / TB, TB, 0, stream>>>(
            (const float4*)numb, (const float4*)denb, (const float4*)x, (float4*)mxb, n4);
    }
    const int Mtiles = N / 16;
    const float invN = 1.f / (float)N;
    // GEMM1: t1 = h0 @ W1 + b1                         [N,128]
    gemm16_wmma<64, 128, 0, true><<<dim3(128 / 16, Mtiles), 32, 0, stream>>>(
        mxb, W1, t1, b1, nullptr, nullptr);
    stats_kernel<<<512, TB, 0, stream>>>(t1, 128, N, st);
    finalize_kernel<<<1, 128, 0, stream>>>(st, g_mlp, be_mlp, st + 512, st + 640, 128, invN);
    // GEMM2: h2 = relu(bn(t1)) @ W2 + b2               [N,64]  (BN+ReLU fused into A load)
    gemm16_wmma<128, 64, 1, true><<<dim3(64 / 16, Mtiles), 32, 0, stream>>>(
        t1, W2, denb, b2, st + 512, st + 640);
    stats_kernel<<<512, TB, 0, stream>>>(denb, 64, N, st + 256);
    finalize_kernel<<<1, 128, 0, stream>>>(st + 256, g1, be1, st + 768, st + 832, 64, invN);
    // GEMM3: h3 = silu(bn(h2)) @ Wl                    [N,64]  (BN+SiLU fused into A load)
    gemm16_wmma<64, 64, 2, false><<<dim3(64 / 16, Mtiles), 32, 0, stream>>>(
        denb, Wl, numb, nullptr, st + 768, st + 832);
    stats_kernel<<<512, TB, 0, stream>>>(numb, 64, N, st + 384);
    finalize_kernel<<<1, 128, 0, stream>>>(st + 384, g2, be2, st + 896, st + 960, 64, invN);
    // out = silu(bn(h3))
    {
        int total = (int)NC;
        bn_silu_out<<<(total + TB - 1) / TB, TB, 0, stream>>>(numb, st + 896, st + 960, out, total);
    }
}